// SAGE_Mini_39711267619348
// MI455X (gfx1250) — compile-verified
//
#include <hip/hip_runtime.h>
#include <hip/hip_bf16.h>
#include <math.h>

#define N_NODES 100000
#define N_EDGES 1600000
#define IN_DIM  128
#define HID_DIM 128
#define OUT_DIM 47
#define OUT_PAD 48

typedef float v2f __attribute__((ext_vector_type(2)));
typedef float v8f __attribute__((ext_vector_type(8)));

// ---------------- utility kernels ----------------

__global__ void zero_f32(float* __restrict__ p, size_t n) {
  size_t i = (size_t)blockIdx.x * blockDim.x + threadIdx.x;
  size_t stride = (size_t)gridDim.x * blockDim.x;
  for (; i < n; i += stride) p[i] = 0.0f;
}

__global__ void zero_f32x4(float4* __restrict__ p, size_t n4) {
  size_t i = (size_t)blockIdx.x * blockDim.x + threadIdx.x;
  size_t stride = (size_t)gridDim.x * blockDim.x;
  const float4 z = {0.f, 0.f, 0.f, 0.f};
  for (; i < n4; i += stride) p[i] = z;
}

__global__ void deg_count(const int* __restrict__ dst, float* __restrict__ deg, int E) {
  int i = blockIdx.x * blockDim.x + threadIdx.x;
  if (i < E) atomicAdd(&deg[dst[i]], 1.0f);
}

__global__ void invert_deg(float* __restrict__ deg, int N) {
  int i = blockIdx.x * blockDim.x + threadIdx.x;
  if (i < N) {
    float d = deg[i];
    deg[i] = (d > 0.0f) ? (1.0f / d) : 0.0f;
  }
}

// Pad a [rows x 47] row-major matrix to [rows x 48], zero-filling col 47.
__global__ void pad_w47(const float* __restrict__ W, float* __restrict__ Wp, int rows) {
  int i = blockIdx.x * blockDim.x + threadIdx.x;
  if (i >= rows * OUT_PAD) return;
  int r = i / OUT_PAD, c = i % OUT_PAD;
  Wp[i] = (c < OUT_DIM) ? W[r * OUT_DIM + c] : 0.0f;
}

// ---------------- neighbor aggregation (scatter-add) ----------------
// One wave per edge. Each lane reads float4 of the 128-float source row
// (512 B fully coalesced) and atomically adds into the destination row.
__global__ __launch_bounds__(256)
void scatter_agg(const float* __restrict__ h, const int* __restrict__ src,
                 const int* __restrict__ dst, float* __restrict__ agg, int E) {
  int edge = blockIdx.x * (blockDim.x >> 5) + (threadIdx.x >> 5);
  int lane = threadIdx.x & 31;
  if (edge >= E) return;
  int s = src[edge];
  int d = dst[edge];
  const float4* hp = (const float4*)(h + (size_t)s * HID_DIM);
  float4 v = hp[lane];
  float* ap = agg + (size_t)d * HID_DIM + lane * 4;
  atomicAdd(ap + 0, v.x);
  atomicAdd(ap + 1, v.y);
  atomicAdd(ap + 2, v.z);
  atomicAdd(ap + 3, v.w);
}

// ---------------- fused SAGE layer GEMM via fp32 WMMA ----------------
// out[m, n] = sum_k inv_deg[m]*agg[m,k]*Wl[k,n] + bl[n] + sum_k h[m,k]*Wr[k,n]
// One wave computes a 16 x (16*NT) output strip with V_WMMA_F32_16X16X4_F32,
// reusing each A fragment across NT column tiles. No predication inside the
// K-loop: weights are pre-padded so every load is in-bounds; only the final
// store masks ncol < OUTN (for the 47-wide layer).
//   WN   = padded weight/bias column count (128 or 48)
//   NT   = number of 16-wide column tiles per wave
//   OUTN = output store stride / true column count (128 or 47)
template<int WN, int NT, bool RELU, int OUTN>
__global__ __launch_bounds__(32)
void sage_gemm_wmma(const float* __restrict__ agg, const float* __restrict__ h,
                    const float* __restrict__ inv_deg,
                    const float* __restrict__ Wl, const float* __restrict__ bl,
                    const float* __restrict__ Wr, float* __restrict__ out) {
  const int m0   = blockIdx.x * 16;
  const int n0   = blockIdx.y * (16 * NT);
  const int lane = threadIdx.x;
  const int l15  = lane & 15;
  const int hi   = lane >> 4;     // 0 or 1 (half-wave)
  const int koff = hi * 2;        // K sub-offset per 16x4 / 4x16 VGPR layout
  const int mrow = m0 + l15;

  v8f c[NT];
#pragma unroll
  for (int t = 0; t < NT; ++t) c[t] = (v8f){};

  const float scale = inv_deg[mrow];
  const float* Arow = agg + (size_t)mrow * HID_DIM + koff;
  const float* Hrow = h   + (size_t)mrow * HID_DIM + koff;

  // half 1: scaled aggregate @ Wl
#pragma unroll 4
  for (int kk = 0; kk < HID_DIM; kk += 4) {
    const float2 av = *(const float2*)(Arow + kk);
    v2f a; a.x = av.x * scale; a.y = av.y * scale;
    const float* Wp = Wl + (size_t)(kk + koff) * WN + n0 + l15;
#pragma unroll
    for (int t = 0; t < NT; ++t) {
      v2f b; b.x = Wp[t * 16]; b.y = Wp[WN + t * 16];
      c[t] = __builtin_amdgcn_wmma_f32_16x16x4_f32(false, a, false, b,
                                                   (short)0, c[t], false, false);
    }
  }
  // half 2: h @ Wr
#pragma unroll 4
  for (int kk = 0; kk < HID_DIM; kk += 4) {
    const float2 hv = *(const float2*)(Hrow + kk);
    v2f a; a.x = hv.x; a.y = hv.y;
    const float* Wp = Wr + (size_t)(kk + koff) * WN + n0 + l15;
#pragma unroll
    for (int t = 0; t < NT; ++t) {
      v2f b; b.x = Wp[t * 16]; b.y = Wp[WN + t * 16];
      c[t] = __builtin_amdgcn_wmma_f32_16x16x4_f32(false, a, false, b,
                                                   (short)0, c[t], false, false);
    }
  }

  // epilogue: bias, ReLU, store (mask columns only for the 47-wide layer)
#pragma unroll
  for (int t = 0; t < NT; ++t) {
    const int ncol = n0 + t * 16 + l15;
    const float bias = bl[ncol];  // padded bias for the 47-wide layer
#pragma unroll
    for (int i = 0; i < 8; ++i) {
      float v = c[t][i] + bias;
      if (RELU) v = fmaxf(v, 0.0f);
      const int row = m0 + i + hi * 8;
      if (WN == OUTN || ncol < OUTN)
        out[(size_t)row * OUTN + ncol] = v;
    }
  }
}

// ---------------- log_softmax over 47 classes, in place ----------------
__global__ __launch_bounds__(128)
void log_softmax_47(float* __restrict__ out, int N) {
  const int row  = blockIdx.x * (blockDim.x >> 5) + (threadIdx.x >> 5);
  const int lane = threadIdx.x & 31;
  if (row >= N) return;   // uniform per wave
  float* p = out + (size_t)row * OUT_DIM;
  float v0 = (lane < OUT_DIM)      ? p[lane]      : -INFINITY;
  float v1 = (lane + 32 < OUT_DIM) ? p[lane + 32] : -INFINITY;
  float m = fmaxf(v0, v1);
#pragma unroll
  for (int off = 16; off > 0; off >>= 1)
    m = fmaxf(m, __shfl_xor(m, off, 32));
  float s = 0.0f;
  if (lane < OUT_DIM)      s += __expf(v0 - m);
  if (lane + 32 < OUT_DIM) s += __expf(v1 - m);
#pragma unroll
  for (int off = 16; off > 0; off >>= 1)
    s += __shfl_xor(s, off, 32);
  const float lse = m + __logf(s);
  if (lane < OUT_DIM)      p[lane]      = v0 - lse;
  if (lane + 32 < OUT_DIM) p[lane + 32] = v1 - lse;
}

// ---------------- launch ----------------

extern "C" void kernel_launch(void* const* d_in, const int* in_sizes, int n_in,
                              void* d_out, int out_size, void* d_ws, size_t ws_size,
                              hipStream_t stream) {
  const float* x   = (const float*)d_in[0];
  const int*   src = (const int*)  d_in[1];
  const int*   dst = (const int*)  d_in[2];
  const float* Wl0 = (const float*)d_in[3];
  const float* bl0 = (const float*)d_in[4];
  const float* Wr0 = (const float*)d_in[5];
  const float* Wl1 = (const float*)d_in[6];
  const float* bl1 = (const float*)d_in[7];
  const float* Wr1 = (const float*)d_in[8];
  const float* Wl2 = (const float*)d_in[9];
  const float* bl2 = (const float*)d_in[10];
  const float* Wr2 = (const float*)d_in[11];
  float* out = (float*)d_out;

  char* ws = (char*)d_ws;
  const size_t matElems = (size_t)N_NODES * HID_DIM;
  float* inv_deg = (float*)ws;                                 // N floats (400000 B, 16B-aligned)
  float* agg  = (float*)(ws + (size_t)N_NODES * sizeof(float));
  float* hA   = agg + matElems;
  float* hB   = hA + matElems;
  float* Wl2p = hB + matElems;                 // 128 x 48
  float* Wr2p = Wl2p + HID_DIM * OUT_PAD;      // 128 x 48
  float* bl2p = Wr2p + HID_DIM * OUT_PAD;      // 48

  // degree -> inverse degree (in place)
  zero_f32<<<512, 256, 0, stream>>>(inv_deg, (size_t)N_NODES);
  deg_count<<<(N_EDGES + 255) / 256, 256, 0, stream>>>(dst, inv_deg, N_EDGES);
  invert_deg<<<(N_NODES + 255) / 256, 256, 0, stream>>>(inv_deg, N_NODES);

  // zero-pad layer-2 weights/bias to 48 columns
  pad_w47<<<(HID_DIM * OUT_PAD + 255) / 256, 256, 0, stream>>>(Wl2, Wl2p, HID_DIM);
  pad_w47<<<(HID_DIM * OUT_PAD + 255) / 256, 256, 0, stream>>>(Wr2, Wr2p, HID_DIM);
  pad_w47<<<1, 64, 0, stream>>>(bl2, bl2p, 1);

  const dim3 gemm_blk(32);
  const dim3 grid128(N_NODES / 16, HID_DIM / (16 * 4));   // NT=4 -> grid.y = 2
  const dim3 grid47(N_NODES / 16, 1);                     // NT=3 covers 48 cols
  const int scatter_grid = (N_EDGES + 7) / 8;             // 8 edges / 256-thread block

  // layer 0: agg(x) ; hA = relu(agg@Wl0 + bl0 + x@Wr0)
  zero_f32x4<<<2048, 256, 0, stream>>>((float4*)agg, matElems / 4);
  scatter_agg<<<scatter_grid, 256, 0, stream>>>(x, src, dst, agg, N_EDGES);
  sage_gemm_wmma<HID_DIM, 4, true, HID_DIM><<<grid128, gemm_blk, 0, stream>>>(
      agg, x, inv_deg, Wl0, bl0, Wr0, hA);

  // layer 1
  zero_f32x4<<<2048, 256, 0, stream>>>((float4*)agg, matElems / 4);
  scatter_agg<<<scatter_grid, 256, 0, stream>>>(hA, src, dst, agg, N_EDGES);
  sage_gemm_wmma<HID_DIM, 4, true, HID_DIM><<<grid128, gemm_blk, 0, stream>>>(
      agg, hA, inv_deg, Wl1, bl1, Wr1, hB);

  // layer 2 (no relu), padded weights, logits straight into d_out (stride 47)
  zero_f32x4<<<2048, 256, 0, stream>>>((float4*)agg, matElems / 4);
  scatter_agg<<<scatter_grid, 256, 0, stream>>>(hB, src, dst, agg, N_EDGES);
  sage_gemm_wmma<OUT_PAD, 3, false, OUT_DIM><<<grid47, gemm_blk, 0, stream>>>(
      agg, hB, inv_deg, Wl2p, bl2p, Wr2p, out);

  // log_softmax in place on d_out
  log_softmax_47<<<(N_NODES + 3) / 4, 128, 0, stream>>>(out, N_NODES);
}